// RWKV_34815004901976
// MI455X (gfx1250) — compile-verified
//
#include <hip/hip_runtime.h>
#include <math.h>

#define DDIM 2048
#define HDIM 8192
#define LNUM 6
#define VDIM 50257

typedef float v2f __attribute__((ext_vector_type(2)));
typedef float v4f __attribute__((ext_vector_type(4)));
typedef float v8f __attribute__((ext_vector_type(8)));
typedef int   gv4i __attribute__((vector_size(4 * sizeof(int))));

#if defined(__has_builtin)
#  if __has_builtin(__builtin_amdgcn_global_load_async_to_lds_b128)
#    define HAVE_ASYNC_LDS 1
#  else
#    define HAVE_ASYNC_LDS 0
#  endif
#else
#  define HAVE_ASYNC_LDS 0
#endif

// ---------------------------------------------------------------------------
// GEMV via V_WMMA_F32_16X16X4_F32 (full fp32 — matches reference precision)
// y[m] = epilogue( sum_k W[m,k] * x[k] ),  K compile-time (2048 or 8192).
//
// Per-wave 16-row tile. A-fragment layout (ISA 16x4 f32):
//   lane l (l&15 = row M), half = l>>4:
//     VGPR0 = W[M, k0+2*half], VGPR1 = W[M, k0+2*half+1]
// B = x broadcast into all 16 N columns -> every D column equals the y-tile.
// x staged into LDS once per block (GLOBAL_LOAD_ASYNC_TO_LDS_B128 when the
// toolchain exposes the gfx1250 builtin, else float4 loads); B reads are
// half-wave broadcasts (conflict-free ds loads).
// MODE: 0=acc*scale  1=sigmoid(acc)  2=relu(acc)^2
//       3: y = (y + (aux?aux[m]:1)*acc)*scale
// ---------------------------------------------------------------------------
template <int MODE, int K>
__global__ __launch_bounds__(256) void gemv_wmma_kernel(
    const float* __restrict__ W, const float* __restrict__ x,
    float* __restrict__ y, const float* __restrict__ aux,
    int M, float scale)
{
    __shared__ float xs[K];

    const int tid  = threadIdx.x;
    const int lane = tid & 31;
    const int wave = tid >> 5;

    // ---- Stage x into LDS (all waves participate BEFORE any early exit) ----
#if HAVE_ASYNC_LDS
    {
        const __attribute__((address_space(1))) char* gsrc =
            (const __attribute__((address_space(1))) char*)x;
        __attribute__((address_space(3))) char* ldst =
            (__attribute__((address_space(3))) char*)xs;
        // 16B per lane per call; K*4 bytes total, uniform trip count.
        for (int i = tid * 16; i < K * 4; i += 256 * 16) {
            __builtin_amdgcn_global_load_async_to_lds_b128(
                (__attribute__((address_space(1))) gv4i*)(gsrc + i),
                (__attribute__((address_space(3))) gv4i*)(ldst + i),
                0, 0);
        }
#  if __has_builtin(__builtin_amdgcn_s_wait_asynccnt)
        __builtin_amdgcn_s_wait_asynccnt(0);
#  else
        asm volatile("s_wait_asynccnt 0x0" ::: "memory");
#  endif
    }
#else
    {
        const v4f* __restrict__ xv4 = (const v4f*)x;
        v4f* xs4 = (v4f*)xs;
        for (int i = tid; i < K / 4; i += 256) xs4[i] = xv4[i];   // b128 pairs
    }
#endif
    __syncthreads();

    const int m0 = (blockIdx.x * 8 + wave) << 4;   // 8 waves/block * 16 rows
    if (m0 >= M) return;                           // wave-uniform: EXEC all-1s

    const int half = lane >> 4;                    // 0 or 1
    int row = m0 + (lane & 15);
    if (row >= M) row = M - 1;                     // clamp loads; store guarded
    const float* Wr = W + (size_t)row * K + (half << 1);

    v8f c0 = {}, c1 = {}, c2 = {}, c3 = {};
    for (int k0 = 0; k0 < K; k0 += 16) {
        if (k0 + 1024 < K)                          // stay inside this row
            __builtin_prefetch(Wr + k0 + 1024, 0, 2);  // global_prefetch_b8

        v2f a0 = *reinterpret_cast<const v2f*>(Wr + k0);
        v2f a1 = *reinterpret_cast<const v2f*>(Wr + k0 + 4);
        v2f a2 = *reinterpret_cast<const v2f*>(Wr + k0 + 8);
        v2f a3 = *reinterpret_cast<const v2f*>(Wr + k0 + 12);

        const int kb = k0 + (half << 1);
        v2f b0, b1, b2, b3;
        b0[0] = xs[kb];      b0[1] = xs[kb + 1];
        b1[0] = xs[kb + 4];  b1[1] = xs[kb + 5];
        b2[0] = xs[kb + 8];  b2[1] = xs[kb + 9];
        b3[0] = xs[kb + 12]; b3[1] = xs[kb + 13];

        c0 = __builtin_amdgcn_wmma_f32_16x16x4_f32(false, a0, false, b0,
                                                   (short)0, c0, false, false);
        c1 = __builtin_amdgcn_wmma_f32_16x16x4_f32(false, a1, false, b1,
                                                   (short)0, c1, false, false);
        c2 = __builtin_amdgcn_wmma_f32_16x16x4_f32(false, a2, false, b2,
                                                   (short)0, c2, false, false);
        c3 = __builtin_amdgcn_wmma_f32_16x16x4_f32(false, a3, false, b3,
                                                   (short)0, c3, false, false);
    }
    c0 = (c0 + c1) + (c2 + c3);

    // D layout: lane l, VGPR v -> row M = v + 8*(l>>4), col N = l&15.
    // All N columns identical; (l&15)==0 lanes (0 and 16) write the 16 rows.
    if ((lane & 15) == 0) {
#pragma unroll
        for (int v = 0; v < 8; ++v) {
            int m = m0 + (half << 3) + v;
            if (m < M) {
                float acc = c0[v];
                if (MODE == 0)      y[m] = acc * scale;
                else if (MODE == 1) y[m] = 1.0f / (1.0f + expf(-acc));
                else if (MODE == 2) { float t = acc > 0.0f ? acc : 0.0f; y[m] = t * t; }
                else { float g = aux ? aux[m] : 1.0f; y[m] = (y[m] + g * acc) * scale; }
            }
        }
    }
}

// ---------------------------------------------------------------------------
// LayerNorm over n elements (single block)
// ---------------------------------------------------------------------------
__global__ __launch_bounds__(256) void ln_kernel(
    const float* __restrict__ x, const float* __restrict__ w,
    const float* __restrict__ b, float* __restrict__ out, int n)
{
    __shared__ float s1[256];
    __shared__ float s2[256];
    const int tid = threadIdx.x;
    float sum = 0.0f, sumsq = 0.0f;
    for (int i = tid; i < n; i += 256) {
        float v = x[i];
        sum += v;
        sumsq += v * v;
    }
    s1[tid] = sum; s2[tid] = sumsq;
    __syncthreads();
    for (int s = 128; s > 0; s >>= 1) {
        if (tid < s) { s1[tid] += s1[tid + s]; s2[tid] += s2[tid + s]; }
        __syncthreads();
    }
    const float mu  = s1[0] / (float)n;
    const float var = s2[0] / (float)n - mu * mu;
    const float inv = rsqrtf(var + 1e-5f);
    for (int i = tid; i < n; i += 256)
        out[i] = (x[i] - mu) * inv * w[i] + b[i];
}

// time-mix: xk/xv/xr = xln*tm + xx*(1-tm)
__global__ void mix3_kernel(const float* __restrict__ xln, const float* __restrict__ xx,
                            const float* __restrict__ tmk, const float* __restrict__ tmv,
                            const float* __restrict__ tmr,
                            float* __restrict__ xk, float* __restrict__ xv,
                            float* __restrict__ xr)
{
    int i = blockIdx.x * blockDim.x + threadIdx.x;
    if (i < DDIM) {
        float a = xln[i], o = xx[i];
        float k = tmk[i], v = tmv[i], r = tmr[i];
        xk[i] = a * k + o * (1.0f - k);
        xv[i] = a * v + o * (1.0f - v);
        xr[i] = a * r + o * (1.0f - r);
    }
}

__global__ void mix2_kernel(const float* __restrict__ xln, const float* __restrict__ xx,
                            const float* __restrict__ tmk, const float* __restrict__ tmr,
                            float* __restrict__ xk, float* __restrict__ xr)
{
    int i = blockIdx.x * blockDim.x + threadIdx.x;
    if (i < DDIM) {
        float a = xln[i], o = xx[i];
        float k = tmk[i], r = tmr[i];
        xk[i] = a * k + o * (1.0f - k);
        xr[i] = a * r + o * (1.0f - r);
    }
}

// WKV recurrence (elementwise); writes r*wkv and new state slots 2..4
__global__ void wkv_kernel(const float* __restrict__ kk, const float* __restrict__ vv,
                           const float* __restrict__ r,  const float* __restrict__ tf,
                           const float* __restrict__ td,
                           const float* __restrict__ aa, const float* __restrict__ bb,
                           const float* __restrict__ pp,
                           float* __restrict__ rwkv, float* __restrict__ naa,
                           float* __restrict__ nbb,  float* __restrict__ np)
{
    int i = blockIdx.x * blockDim.x + threadIdx.x;
    if (i < DDIM) {
        float k = kk[i], v = vv[i];
        float A = aa[i], B = bb[i], P = pp[i];
        float ww = tf[i] + k;
        float p  = fmaxf(P, ww);
        float e1 = expf(P - p), e2 = expf(ww - p);
        float wkv = (e1 * A + e2 * v) / (e1 * B + e2);
        rwkv[i] = r[i] * wkv;
        float ww2 = P + td[i];
        float p2  = fmaxf(ww2, k);
        float e1b = expf(ww2 - p2), e2b = expf(k - p2);
        naa[i] = e1b * A + e2b * v;
        nbb[i] = e1b * B + e2b;
        np[i]  = p2;
    }
}

__global__ void copy_kernel(const float* __restrict__ src, float* __restrict__ dst, int n)
{
    int i = blockIdx.x * blockDim.x + threadIdx.x;
    if (i < n) dst[i] = src[i];
}

// ---------------------------------------------------------------------------
extern "C" void kernel_launch(void* const* d_in, const int* in_sizes, int n_in,
                              void* d_out, int out_size, void* d_ws, size_t ws_size,
                              hipStream_t stream)
{
    (void)in_sizes; (void)n_in; (void)out_size; (void)ws_size;

    const float* token_embd = (const float*)d_in[0];
    const float* state      = (const float*)d_in[1];
    const float* ln0_w      = (const float*)d_in[2];
    const float* ln0_b      = (const float*)d_in[3];
    const float* ln1_w      = (const float*)d_in[4];
    const float* ln1_b      = (const float*)d_in[5];
    const float* ln2_w      = (const float*)d_in[6];
    const float* ln2_b      = (const float*)d_in[7];
    const float* att_tmk    = (const float*)d_in[8];
    const float* att_tmv    = (const float*)d_in[9];
    const float* att_tmr    = (const float*)d_in[10];
    const float* att_tf     = (const float*)d_in[11];
    const float* att_td     = (const float*)d_in[12];
    const float* att_kw     = (const float*)d_in[13];
    const float* att_vw     = (const float*)d_in[14];
    const float* att_rw     = (const float*)d_in[15];
    const float* att_ow     = (const float*)d_in[16];
    const float* ffn_tmk    = (const float*)d_in[17];
    const float* ffn_tmr    = (const float*)d_in[18];
    const float* ffn_kw     = (const float*)d_in[19];
    const float* ffn_vw     = (const float*)d_in[20];
    const float* ffn_rw     = (const float*)d_in[21];
    const float* lnout_w    = (const float*)d_in[22];
    const float* lnout_b    = (const float*)d_in[23];
    const float* head_w     = (const float*)d_in[24];

    float* out    = (float*)d_out;
    float* logits = out;               // V floats
    float* ost    = out + VDIM;        // 5L x D new state

    float* ws   = (float*)d_ws;
    float* x    = ws;  ws += DDIM;
    float* xln  = ws;  ws += DDIM;
    float* xk   = ws;  ws += DDIM;
    float* xv   = ws;  ws += DDIM;
    float* xr   = ws;  ws += DDIM;
    float* rr   = ws;  ws += DDIM;
    float* kk   = ws;  ws += DDIM;
    float* vv   = ws;  ws += DDIM;
    float* rwkv = ws;  ws += DDIM;
    float* fxk  = ws;  ws += DDIM;
    float* fxr  = ws;  ws += DDIM;
    float* fr   = ws;  ws += DDIM;
    float* k2   = ws;  ws += HDIM;

    const dim3 eb(256);
    const dim3 eg((DDIM + 255) / 256);
    auto ggrid = [](int M) { return dim3((unsigned)((M + 127) / 128)); };

    // pass-through state slots (slots 2..4 overwritten per layer below)
    copy_kernel<<<(5 * LNUM * DDIM + 255) / 256, 256, 0, stream>>>(state, ost, 5 * LNUM * DDIM);

    // x0 = LN(token_embd)
    ln_kernel<<<1, 256, 0, stream>>>(token_embd, ln0_w, ln0_b, x, DDIM);

    for (int l = 0; l < LNUM; ++l) {
        const size_t dd = (size_t)l * DDIM * DDIM;
        const size_t hd = (size_t)l * HDIM * DDIM;
        const size_t ld = (size_t)l * DDIM;
        const float* s0 = state + (size_t)(l * 5 + 0) * DDIM;
        const float* s1 = state + (size_t)(l * 5 + 1) * DDIM;
        const float* s2 = state + (size_t)(l * 5 + 2) * DDIM;
        const float* s3 = state + (size_t)(l * 5 + 3) * DDIM;
        const float* s4 = state + (size_t)(l * 5 + 4) * DDIM;
        float* naa = ost + (size_t)(l * 5 + 2) * DDIM;
        float* nbb = ost + (size_t)(l * 5 + 3) * DDIM;
        float* np  = ost + (size_t)(l * 5 + 4) * DDIM;

        // ---- time mixing ----
        ln_kernel<<<1, 256, 0, stream>>>(x, ln1_w + ld, ln1_b + ld, xln, DDIM);
        mix3_kernel<<<eg, eb, 0, stream>>>(xln, s1, att_tmk + ld, att_tmv + ld,
                                           att_tmr + ld, xk, xv, xr);
        gemv_wmma_kernel<1, DDIM><<<ggrid(DDIM), 256, 0, stream>>>(att_rw + dd, xr, rr,
                                                                   nullptr, DDIM, 1.0f);
        gemv_wmma_kernel<0, DDIM><<<ggrid(DDIM), 256, 0, stream>>>(att_kw + dd, xk, kk,
                                                                   nullptr, DDIM, 1.0f);
        gemv_wmma_kernel<0, DDIM><<<ggrid(DDIM), 256, 0, stream>>>(att_vw + dd, xv, vv,
                                                                   nullptr, DDIM, 1.0f);
        wkv_kernel<<<eg, eb, 0, stream>>>(kk, vv, rr, att_tf + ld, att_td + ld,
                                          s2, s3, s4, rwkv, naa, nbb, np);
        // x += ow @ (r*wkv)
        gemv_wmma_kernel<3, DDIM><<<ggrid(DDIM), 256, 0, stream>>>(att_ow + dd, rwkv, x,
                                                                   nullptr, DDIM, 1.0f);

        // ---- channel mixing ----
        ln_kernel<<<1, 256, 0, stream>>>(x, ln2_w + ld, ln2_b + ld, xln, DDIM);
        mix2_kernel<<<eg, eb, 0, stream>>>(xln, s0, ffn_tmk + ld, ffn_tmr + ld, fxk, fxr);
        gemv_wmma_kernel<1, DDIM><<<ggrid(DDIM), 256, 0, stream>>>(ffn_rw + dd, fxr, fr,
                                                                   nullptr, DDIM, 1.0f);
        gemv_wmma_kernel<2, DDIM><<<ggrid(HDIM), 256, 0, stream>>>(ffn_kw + hd, fxk, k2,
                                                                   nullptr, HDIM, 1.0f);
        // x = (x + fr * (vw @ k2)) * (layer==5 ? 0.5 : 1)
        const float sc = ((l + 1) % 6 == 0) ? 0.5f : 1.0f;
        gemv_wmma_kernel<3, HDIM><<<ggrid(DDIM), 256, 0, stream>>>(ffn_vw + hd, k2, x,
                                                                   fr, DDIM, sc);
    }

    // final LN + head
    ln_kernel<<<1, 256, 0, stream>>>(x, lnout_w, lnout_b, xln, DDIM);
    gemv_wmma_kernel<0, DDIM><<<ggrid(VDIM), 256, 0, stream>>>(head_w, xln, logits,
                                                               nullptr, VDIM, 1.0f);
}